// Motif_Loss_30640296689664
// MI455X (gfx1250) — compile-verified
//
#include <hip/hip_runtime.h>
#include <math.h>

#define BATCH 1024
#define MDIM  128
#define DDIM  256
#define KCHUNK 64
#define LDSTRIDE 68   // 64 + 4 floats pad: row stride mod 64 == 4 -> conflict-free tile reads

typedef float v2f __attribute__((ext_vector_type(2)));
typedef float v8f __attribute__((ext_vector_type(8)));

// One block (256 threads = 8 waves) per batch sample.
// Computes per-sample masked MSE((cos-sim - I)) * mask / m^2 into ws[b].
__global__ __launch_bounds__(256)
void motif_gram_kernel(const float* __restrict__ X,
                       const int* __restrict__ motif_num,
                       float* __restrict__ ws) {
    extern __shared__ float smem[];
    float* Xs      = smem;                    // [128][LDSTRIDE]
    float* norms   = Xs + MDIM * LDSTRIDE;    // [128]
    float* partial = norms + MDIM;            // [1]

    const int b    = blockIdx.x;
    const int tid  = threadIdx.x;
    const int lane = tid & 31;
    const int wave = tid >> 5;                // tile-row band i = 0..7
    const int hi   = lane >> 4;               // K sub-pair select (ISA A/B layout)
    const int ln   = lane & 15;               // M (for A) / N (for B,C)

    if (tid == 0) partial[0] = 0.0f;

    const int mnum = motif_num[b];
    const float* __restrict__ Xb = X + (size_t)b * MDIM * DDIM;

    float ss = 0.0f;                          // row-norm^2 accumulator (tid<128 owns row tid)
    v8f cacc[8];
    #pragma unroll
    for (int j = 0; j < 8; ++j) cacc[j] = (v8f){};

    for (int p = 0; p < 4; ++p) {
        // ---- stage K-chunk p into LDS: 128 rows x 64 floats, float4 granularity ----
        #pragma unroll
        for (int it = 0; it < 8; ++it) {
            int q   = tid + 256 * it;         // float4 index within chunk
            int row = q >> 4;                 // 16 float4 per row
            int c4  = q & 15;
            float4 v = *(const float4*)(Xb + row * DDIM + p * KCHUNK + c4 * 4);
            *(float4*)(Xs + row * LDSTRIDE + c4 * 4) = v;
        }
        __syncthreads();

        // ---- accumulate row-norm^2 for this chunk ----
        if (tid < MDIM) {
            const float* r = Xs + tid * LDSTRIDE;
            #pragma unroll
            for (int k = 0; k < KCHUNK; k += 4) {
                float4 v = *(const float4*)(r + k);
                ss += v.x * v.x + v.y * v.y + v.z * v.z + v.w * v.w;
            }
        }

        // ---- Gram tiles via fp32 WMMA: this wave computes tiles (wave, j=0..7) ----
        // A 16x4 layout: lane ln holds row M=wave*16+ln; VGPR pair holds K = 2*hi, 2*hi+1
        const float* arow = Xs + (wave * 16 + ln) * LDSTRIDE + 2 * hi;
        #pragma unroll
        for (int j = 0; j < 8; ++j) {
            // B 4x16 = X^T tile: lane ln supplies column n -> row (j*16+ln) of X
            const float* brow = Xs + (j * 16 + ln) * LDSTRIDE + 2 * hi;
            v8f c = cacc[j];
            #pragma unroll
            for (int k = 0; k < KCHUNK; k += 4) {
                v2f a  = *(const v2f*)(arow + k);
                v2f bv = *(const v2f*)(brow + k);
                c = __builtin_amdgcn_wmma_f32_16x16x4_f32(
                        /*neg_a=*/false, a, /*neg_b=*/false, bv,
                        /*c_mod=*/(short)0, c, /*reuse_a=*/false, /*reuse_b=*/false);
            }
            cacc[j] = c;
        }
        __syncthreads();  // before next chunk overwrites Xs
    }

    if (tid < MDIM) norms[tid] = sqrtf(ss);
    __syncthreads();

    // ---- epilogue: sim = dot / max(ni*nj, eps); masked (sim - I)^2 ----
    float lacc = 0.0f;
    #pragma unroll
    for (int j = 0; j < 8; ++j) {
        int   nj  = j * 16 + ln;              // C layout: N = lane%16
        float nnj = norms[nj];
        bool  cv  = nj < mnum;
        #pragma unroll
        for (int r = 0; r < 8; ++r) {
            int   mi    = wave * 16 + hi * 8 + r;   // C layout: M = vgpr + 8*(lane/16)
            float denom = fmaxf(norms[mi] * nnj, 1e-8f);
            float s     = cacc[j][r] / denom;
            if (mi == nj) s -= 1.0f;
            if (cv && (mi < mnum)) lacc += s * s;
        }
    }

    // wave32 tree reduce, then LDS atomic across the 8 waves
    #pragma unroll
    for (int off = 16; off >= 1; off >>= 1)
        lacc += __shfl_down(lacc, off, 32);
    if (lane == 0) atomicAdd(partial, lacc);
    __syncthreads();

    if (tid == 0) {
        float m = (float)mnum;
        ws[b] = partial[0] / (m * m);
    }
}

// Mean over the 1024 per-sample values.
__global__ __launch_bounds__(256)
void motif_mean_kernel(const float* __restrict__ ws, float* __restrict__ out) {
    __shared__ float s[256];
    int t = threadIdx.x;
    float v = ws[t] + ws[t + 256] + ws[t + 512] + ws[t + 768];
    s[t] = v;
    __syncthreads();
    for (int off = 128; off > 0; off >>= 1) {
        if (t < off) s[t] += s[t + off];
        __syncthreads();
    }
    if (t == 0) out[0] = s[0] / (float)BATCH;
}

extern "C" void kernel_launch(void* const* d_in, const int* in_sizes, int n_in,
                              void* d_out, int out_size, void* d_ws, size_t ws_size,
                              hipStream_t stream) {
    const float* X  = (const float*)d_in[0];   // [B, M, D] fp32
    const int*   mn = (const int*)d_in[1];     // [B] int
    float*       ws = (float*)d_ws;            // B floats scratch
    float*       out = (float*)d_out;          // scalar fp32

    size_t smem = (size_t)(MDIM * LDSTRIDE + MDIM + 1) * sizeof(float);
    motif_gram_kernel<<<BATCH, 256, smem, stream>>>(X, mn, ws);
    motif_mean_kernel<<<1, 256, 0, stream>>>(ws, out);
}